// Attention_377957122251
// MI455X (gfx1250) — compile-verified
//
#include <hip/hip_runtime.h>
#include <float.h>

#define B_ 512
#define N_ 1024
#define D_ 256

typedef __attribute__((ext_vector_type(2))) float v2f;
typedef __attribute__((ext_vector_type(8))) float v8f;

// One workgroup (8 wave32) per batch entry.
__global__ __launch_bounds__(256) void attn_mention_select_kernel(
    const float* __restrict__ node,   // [B, N, D] f32
    const int*   __restrict__ edge,   // [B, N]    i32
    const int*   __restrict__ label,  // [B]       i32
    const float* __restrict__ rel,    // [R, D]    f32
    float*       __restrict__ out)    // [B, D]    f32
{
    __shared__ float s_q[D_];
    __shared__ float s_scores[N_];
    __shared__ float s_redmax[8];
    __shared__ float s_redsum[8];

    const int b    = blockIdx.x;
    const int tid  = threadIdx.x;      // 0..255
    const int lane = tid & 31;
    const int wave = tid >> 5;         // 0..7

    // q = relation_matrix[relation_label[b]]  -> LDS (uniform per block)
    s_q[tid] = rel[(size_t)label[b] * D_ + tid];
    __syncthreads();

    const float* nodeB = node + (size_t)b * N_ * D_;

    // -------- Pass 1: scores[n] = dot(node[b,n,:], q) via V_WMMA_F32_16X16X4_F32
    // A (16x4 f32): lane l -> row (l%16), K = (l<16?0:2)+{0,1}   (2 VGPRs)
    // B (4x16 f32): column-broadcast of q chunk -> every D column = score vector
    const int m     = lane & 15;         // A row within tile
    const int khalf = (lane >> 4) << 1;  // 0 (lanes 0-15) or 2 (lanes 16-31)

    for (int t = wave; t < N_ / 16; t += 8) {
        const int row0 = t << 4;
        const float* arow = nodeB + (size_t)(row0 + m) * D_ + khalf;
        const float* brow = s_q + khalf;
        v8f acc = {0.f, 0.f, 0.f, 0.f, 0.f, 0.f, 0.f, 0.f};
#pragma unroll 8
        for (int k0 = 0; k0 < D_; k0 += 4) {
            v2f a, bq;
            a.x  = arow[k0];         // global_load_b64 (contiguous pair)
            a.y  = arow[k0 + 1];
            bq.x = brow[k0];         // ds_load_b64 (16-lane broadcast)
            bq.y = brow[k0 + 1];
            acc = __builtin_amdgcn_wmma_f32_16x16x4_f32(
                      false, a, false, bq, (short)0, acc, false, false);
        }
        // C/D layout: VGPR r holds M=r (lanes 0-15) / M=r+8 (lanes 16-31),
        // replicated across all 16 N-columns -> lanes 0 and 16 spill 8 rows each.
        if ((lane & 15) == 0) {
            const int rbase = row0 + ((lane >> 4) << 3);
#pragma unroll
            for (int r = 0; r < 8; ++r) s_scores[rbase + r] = acc[r];
        }
    }
    __syncthreads();

    // -------- Mask (coalesced edge reads): match jnp.finfo(f32).min semantics
#pragma unroll
    for (int i = tid; i < N_; i += 256) {
        if (edge[(size_t)b * N_ + i] != 1) s_scores[i] = -FLT_MAX;
    }
    __syncthreads();

    // -------- Max reduction (wave32 shuffles + LDS combine)
    float mx = -FLT_MAX;
#pragma unroll
    for (int i = tid; i < N_; i += 256) mx = fmaxf(mx, s_scores[i]);
#pragma unroll
    for (int off = 16; off > 0; off >>= 1)
        mx = fmaxf(mx, __shfl_xor(mx, off, 32));
    if (lane == 0) s_redmax[wave] = mx;
    __syncthreads();
    mx = s_redmax[0];
#pragma unroll
    for (int w = 1; w < 8; ++w) mx = fmaxf(mx, s_redmax[w]);

    // -------- exp + sum reduction; p values back into LDS
    float sum = 0.f;
#pragma unroll
    for (int i = tid; i < N_; i += 256) {
        const float p = __expf(s_scores[i] - mx);
        s_scores[i] = p;
        sum += p;
    }
#pragma unroll
    for (int off = 16; off > 0; off >>= 1)
        sum += __shfl_xor(sum, off, 32);
    if (lane == 0) s_redsum[wave] = sum;
    __syncthreads();
    sum = 0.f;
#pragma unroll
    for (int w = 0; w < 8; ++w) sum += s_redsum[w];
    const float inv = 1.0f / sum;

    // -------- Pass 2: out[b,d] = (1/sum) * sum_n p[n] * node[b,n,d]
    // Thread d owns column d: each n is a fully coalesced 1KB row read (L2 hit),
    // p[n] broadcast from LDS. 4 accumulators for ILP.
    const float* col = nodeB + tid;
    float a0 = 0.f, a1 = 0.f, a2 = 0.f, a3 = 0.f;
    for (int n = 0; n < N_; n += 4) {
        a0 = fmaf(s_scores[n + 0], col[(size_t)(n + 0) * D_], a0);
        a1 = fmaf(s_scores[n + 1], col[(size_t)(n + 1) * D_], a1);
        a2 = fmaf(s_scores[n + 2], col[(size_t)(n + 2) * D_], a2);
        a3 = fmaf(s_scores[n + 3], col[(size_t)(n + 3) * D_], a3);
    }
    out[(size_t)b * D_ + tid] = (a0 + a1 + a2 + a3) * inv;
}

extern "C" void kernel_launch(void* const* d_in, const int* in_sizes, int n_in,
                              void* d_out, int out_size, void* d_ws, size_t ws_size,
                              hipStream_t stream) {
    const float* node  = (const float*)d_in[0];   // [B,N,D] f32
    const int*   edge  = (const int*)  d_in[1];   // [B,N]   i32
    const int*   label = (const int*)  d_in[2];   // [B]     i32
    const float* rel   = (const float*)d_in[3];   // [R,D]   f32
    float*       outp  = (float*)d_out;           // [B,D]   f32
    (void)in_sizes; (void)n_in; (void)out_size; (void)d_ws; (void)ws_size;

    attn_mention_select_kernel<<<dim3(B_), dim3(256), 0, stream>>>(
        node, edge, label, rel, outp);
}